// getllh_53523882443229
// MI455X (gfx1250) — compile-verified
//
#include <hip/hip_runtime.h>
#include <hip/hip_bf16.h>

typedef __attribute__((ext_vector_type(2))) float v2f;
typedef __attribute__((ext_vector_type(8))) float v8f;

#define BB 64
#define TT 2048
#define SS 32
#define KK 8
#define DD 64
#define MROWS (BB * TT)          // 131072
#define NCOLS (SS * KK)          // 256
#define KDIM  128                // [x^2 | x] stacked
#define LOG2PI 1.8378770664093453f
#define LOG2E  1.4426950408889634f
#define INV_XSIZE (1.0f / 8388608.0f)   // 1/(B*T*D)

// ---- workspace layout (floats) ----
#define WS_BMAT    0                         // 128*256 (pair-interleaved)
#define WS_QOFF    (WS_BMAT + KDIM * NCOLS)  // 256
#define WS_LLH     (WS_QOFF + NCOLS)         // B*T*S
#define WS_INNER   (WS_LLH + MROWS * SS)
#define WS_FWD     (WS_INNER + MROWS * SS)
#define WS_BWD     (WS_FWD + MROWS * SS)
#define WS_PART    (WS_BWD + MROWS * SS)     // 2 * NBLK_C
#define NBLK_C     (MROWS / 8)               // 16384 loss-reduction blocks

static __device__ __forceinline__ float fast_exp2(float x) {
#if __has_builtin(__builtin_amdgcn_exp2f)
  return __builtin_amdgcn_exp2f(x);          // v_exp_f32 (base-2 native)
#else
  return exp2f(x);
#endif
}
static __device__ __forceinline__ float fast_log2(float x) {
#if __has_builtin(__builtin_amdgcn_logf)
  return __builtin_amdgcn_logf(x);           // v_log_f32 (base-2 native)
#else
  return __log2f(x);
#endif
}

// -------------------------------------------------------------------------
// Kernel 0: build B' (128 x 256) pair-interleaved so a (k,k+1) fragment for
// one column is a contiguous 8B load; q_off[sk] = logPI - 0.5*cs/xsize
// Pair layout: Bp[(k>>1)*512 + col*2 + (k&1)]
// -------------------------------------------------------------------------
__global__ __launch_bounds__(256) void prep_kernel(
    const float* __restrict__ means, const float* __restrict__ logvars,
    const float* __restrict__ logPIk, float* __restrict__ ws) {
  float* Bm = ws + WS_BMAT;
  float* qoff = ws + WS_QOFF;
  int sk = threadIdx.x;               // 0..255
  const float* mrow = means + sk * DD;
  const float* lrow = logvars + sk * DD;
  float cs = 0.0f;
  for (int d = 0; d < DD; ++d) {
    float lv = lrow[d];
    float p = __expf(-lv);
    float mm = mrow[d];
    int kq = d;                        // quad row
    int kc = DD + d;                   // cross row
    Bm[(kq >> 1) * (NCOLS * 2) + sk * 2 + (kq & 1)] = -0.5f * p;
    Bm[(kc >> 1) * (NCOLS * 2) + sk * 2 + (kc & 1)] = mm * p;
    cs += mm * mm * p + lv + LOG2PI;
  }
  qoff[sk] = logPIk[sk] - 0.5f * cs * INV_XSIZE;
}

// -------------------------------------------------------------------------
// Kernel 1: WMMA GEMM (M=131072, N=256, K=128) fused with per-(b,t,s)
// epilogue producing llh[bt,s] = sum_k q and inner[bt,s] = sum_k softmax(q)*q
// Block tile: 32 rows x 64 cols; 8 waves as 2(M) x 4(N) of 16x16 WMMA tiles.
// Two accumulators break the WMMA C-dependency chain.
// -------------------------------------------------------------------------
template <bool SQ>
static __device__ __forceinline__ v8f wmma_step(const float* __restrict__ xrow,
                                                const v2f* __restrict__ Bp,
                                                int k0, int colg, v8f acc) {
  v2f xv = *(const v2f*)(xrow + (SQ ? k0 : (k0 - 64)));   // 8B-aligned pair
  v2f a = SQ ? (xv * xv) : xv;
  v2f b = Bp[(k0 >> 1) * NCOLS + colg];                   // global_load_b64
  return __builtin_amdgcn_wmma_f32_16x16x4_f32(false, a, false, b,
                                               (short)0, acc, false, false);
}

__global__ __launch_bounds__(256) void loglh_wmma_kernel(
    const float* __restrict__ x, float* __restrict__ ws) {
  const v2f* Bp = (const v2f*)(ws + WS_BMAT);
  const float* qoff = ws + WS_QOFF;
  float* llh = ws + WS_LLH;
  float* innr = ws + WS_INNER;

  __shared__ __align__(16) float xs[32 * 66];  // 32 rows, pad=2 (keeps pairs 8B aligned)

  const int tid = threadIdx.x;
  const int wave = tid >> 5;
  const int lane = tid & 31;
  const int bm = blockIdx.x >> 2;       // 0..4095
  const int bn = blockIdx.x & 3;        // 0..3
  const int rowBase = bm * 32;
  const int colBase = bn * 64;

  // stage x tile (coalesced)
  for (int i = tid; i < 32 * 64; i += 256) {
    int r = i >> 6, c = i & 63;
    xs[r * 66 + c] = x[(size_t)(rowBase + r) * DD + c];
  }
  __syncthreads();

  const int half = lane >> 4;           // 0|1
  const int mrow = lane & 15;
  const int wm = wave >> 2;             // 0..1
  const int wn = wave & 3;              // 0..3
  const float* xrow = &xs[(wm * 16 + mrow) * 66];
  const int colg = colBase + wn * 16 + (lane & 15);

  v8f acc0 = {}, acc1 = {};
  // first 64 K-values: A = x^2
  #pragma unroll
  for (int ks = 0; ks < 16; ks += 2) {
    acc0 = wmma_step<true>(xrow, Bp, ks * 4 + half * 2, colg, acc0);
    acc1 = wmma_step<true>(xrow, Bp, (ks + 1) * 4 + half * 2, colg, acc1);
  }
  // second 64 K-values: A = x
  #pragma unroll
  for (int ks = 16; ks < 32; ks += 2) {
    acc0 = wmma_step<false>(xrow, Bp, ks * 4 + half * 2, colg, acc0);
    acc1 = wmma_step<false>(xrow, Bp, (ks + 1) * 4 + half * 2, colg, acc1);
  }
  v8f acc = acc0 + acc1;

  // epilogue: q = raw/xsize + qoff;  reduce over k (groups of 8 lanes)
  const float qo = qoff[colg];
  const int sg = colg >> 3;
  #pragma unroll
  for (int r = 0; r < 8; ++r) {
    int m = r + 8 * half;
    int bt = rowBase + wm * 16 + m;
    float q = acc[r] * INV_XSIZE + qo;

    float mx = q;
    mx = fmaxf(mx, __shfl_xor(mx, 1));
    mx = fmaxf(mx, __shfl_xor(mx, 2));
    mx = fmaxf(mx, __shfl_xor(mx, 4));

    float sq = q;
    sq += __shfl_xor(sq, 1); sq += __shfl_xor(sq, 2); sq += __shfl_xor(sq, 4);

    float e = __expf(q - mx);
    float se = e;
    se += __shfl_xor(se, 1); se += __shfl_xor(se, 2); se += __shfl_xor(se, 4);

    float nu = e * q;
    nu += __shfl_xor(nu, 1); nu += __shfl_xor(nu, 2); nu += __shfl_xor(nu, 4);

    if ((lane & 7) == 0) {
      llh[(size_t)bt * SS + sg] = sq;
      innr[(size_t)bt * SS + sg] = nu / se;
    }
  }
}

// -------------------------------------------------------------------------
// Kernel 2: forward + backward HMM recursion in the log2 domain (v_exp_f32 /
// v_log_f32 are natively base-2). 1 block per batch, 2 waves: wave 0 =
// forward (lane = state j), wave 1 = backward (lane = state i). The (scaled)
// transition column/row lives in 32 registers per lane.
// fwd/bwd outputs are stored scaled by log2(e); softmax in kernel 3 is exact.
// -------------------------------------------------------------------------
__global__ __launch_bounds__(64) void fwdbwd_kernel(
    const float* __restrict__ log_startprob,
    const float* __restrict__ log_transmat, float* __restrict__ ws) {
  __shared__ float Mt[32 * 32];    // M[i][j] * LOG2E
  __shared__ float Mtt[32 * 32];   // transposed: Mtt[j][i]

  const int b = blockIdx.x;
  const int tid = threadIdx.x;
  for (int i = tid; i < 1024; i += 64) {
    float v = log_transmat[i] * LOG2E;
    Mt[i] = v;
    Mtt[(i & 31) * 32 + (i >> 5)] = v;
  }
  __syncthreads();

  const int lane = tid & 31;
  const float* lptr = ws + WS_LLH + (size_t)b * TT * SS;

  float Mreg[32];
  if (tid < 32) {
    // ---------------- forward ----------------
    #pragma unroll
    for (int i = 0; i < 32; ++i) Mreg[i] = Mt[i * 32 + lane];  // column j=lane
    float* fwd = ws + WS_FWD + (size_t)b * TT * SS;
    float a = (log_startprob[lane] + lptr[lane]) * LOG2E;
    fwd[lane] = a;
    for (int t = 1; t < TT; ++t) {
      if (t + 4 < TT) __builtin_prefetch(lptr + (t + 4) * SS + lane, 0, 1);
      float am = a;
      am = fmaxf(am, __shfl_xor(am, 16));
      am = fmaxf(am, __shfl_xor(am, 8));
      am = fmaxf(am, __shfl_xor(am, 4));
      am = fmaxf(am, __shfl_xor(am, 2));
      am = fmaxf(am, __shfl_xor(am, 1));
      float as = a - am;
      float s = 0.0f;
      #pragma unroll
      for (int i = 0; i < 32; ++i)
        s += fast_exp2(__shfl(as, i) + Mreg[i]);
      a = am + fast_log2(s) + lptr[t * SS + lane] * LOG2E;
      fwd[t * SS + lane] = a;
    }
  } else {
    // ---------------- backward ----------------
    #pragma unroll
    for (int j = 0; j < 32; ++j) Mreg[j] = Mtt[j * 32 + lane];  // row i=lane
    float* bwd = ws + WS_BWD + (size_t)b * TT * SS;
    float bv = 0.0f;
    bwd[(TT - 1) * SS + lane] = 0.0f;
    for (int t = TT - 2; t >= 0; --t) {
      if (t >= 4) __builtin_prefetch(lptr + (t - 3) * SS + lane, 0, 1);
      float cv = lptr[(t + 1) * SS + lane] * LOG2E + bv;   // lane = j
      float cm = cv;
      cm = fmaxf(cm, __shfl_xor(cm, 16));
      cm = fmaxf(cm, __shfl_xor(cm, 8));
      cm = fmaxf(cm, __shfl_xor(cm, 4));
      cm = fmaxf(cm, __shfl_xor(cm, 2));
      cm = fmaxf(cm, __shfl_xor(cm, 1));
      float cs = cv - cm;
      float s = 0.0f;
      #pragma unroll
      for (int j = 0; j < 32; ++j)
        s += fast_exp2(__shfl(cs, j) + Mreg[j]);           // M[lane][j]
      bv = cm + fast_log2(s);
      bwd[t * SS + lane] = bv;
    }
  }
}

// -------------------------------------------------------------------------
// Kernel 3: posteriors + loss partials. One wave per (b,t); lane = state.
// fwd/bwd are log2e-scaled: exp2(g' - gmax') == exp(g - gmax), so the
// softmax is exact. Per-block deterministic partial sums (no atomics).
// -------------------------------------------------------------------------
__global__ __launch_bounds__(256) void loss_kernel(float* __restrict__ ws) {
  const float* fwd = ws + WS_FWD;
  const float* bwd = ws + WS_BWD;
  const float* innr = ws + WS_INNER;
  const float* llh = ws + WS_LLH;
  float* part = ws + WS_PART;

  __shared__ float sl[8], sh[8];
  const int wave = threadIdx.x >> 5;
  const int lane = threadIdx.x & 31;
  const size_t bt = (size_t)blockIdx.x * 8 + wave;

  float g = fwd[bt * SS + lane] + bwd[bt * SS + lane];   // log2e-scaled
  float gm = g;
  gm = fmaxf(gm, __shfl_xor(gm, 16));
  gm = fmaxf(gm, __shfl_xor(gm, 8));
  gm = fmaxf(gm, __shfl_xor(gm, 4));
  gm = fmaxf(gm, __shfl_xor(gm, 2));
  gm = fmaxf(gm, __shfl_xor(gm, 1));
  float e = fast_exp2(g - gm);
  float se = e;
  se += __shfl_xor(se, 16); se += __shfl_xor(se, 8);
  se += __shfl_xor(se, 4);  se += __shfl_xor(se, 2); se += __shfl_xor(se, 1);

  float contrib = (e / se) * innr[bt * SS + lane];
  float lv = llh[bt * SS + lane];
  contrib += __shfl_xor(contrib, 16); contrib += __shfl_xor(contrib, 8);
  contrib += __shfl_xor(contrib, 4);  contrib += __shfl_xor(contrib, 2);
  contrib += __shfl_xor(contrib, 1);
  lv += __shfl_xor(lv, 16); lv += __shfl_xor(lv, 8);
  lv += __shfl_xor(lv, 4);  lv += __shfl_xor(lv, 2); lv += __shfl_xor(lv, 1);

  if (lane == 0) { sl[wave] = contrib; sh[wave] = lv; }
  __syncthreads();
  if (threadIdx.x == 0) {
    float s1 = 0.0f, s2 = 0.0f;
    for (int w = 0; w < 8; ++w) { s1 += sl[w]; s2 += sh[w]; }
    part[blockIdx.x] = s1;
    part[NBLK_C + blockIdx.x] = s2;
  }
}

// -------------------------------------------------------------------------
// Kernel 4: final deterministic reduction of partials -> d_out[2]
// -------------------------------------------------------------------------
__global__ __launch_bounds__(256) void final_kernel(const float* __restrict__ ws,
                                                    float* __restrict__ out) {
  const float* part = ws + WS_PART;
  __shared__ float r1[256], r2[256];
  const int tid = threadIdx.x;
  float s1 = 0.0f, s2 = 0.0f;
  for (int i = tid; i < NBLK_C; i += 256) {
    s1 += part[i];
    s2 += part[NBLK_C + i];
  }
  r1[tid] = s1; r2[tid] = s2;
  __syncthreads();
  for (int off = 128; off > 0; off >>= 1) {
    if (tid < off) { r1[tid] += r1[tid + off]; r2[tid] += r2[tid + off]; }
    __syncthreads();
  }
  if (tid == 0) {
    out[0] = -r1[0] / (float)(TT * BB);   // loss.sum()
    out[1] = r2[0];                       // llh.sum()
  }
}

// -------------------------------------------------------------------------
extern "C" void kernel_launch(void* const* d_in, const int* in_sizes, int n_in,
                              void* d_out, int out_size, void* d_ws, size_t ws_size,
                              hipStream_t stream) {
  const float* x = (const float*)d_in[0];
  const float* means = (const float*)d_in[1];
  const float* logvars = (const float*)d_in[2];
  const float* logPIk_s = (const float*)d_in[3];
  const float* log_startprob = (const float*)d_in[4];
  const float* log_transmat = (const float*)d_in[5];
  float* out = (float*)d_out;
  float* ws = (float*)d_ws;

  prep_kernel<<<1, 256, 0, stream>>>(means, logvars, logPIk_s, ws);
  loglh_wmma_kernel<<<(MROWS / 32) * 4, 256, 0, stream>>>(x, ws);
  fwdbwd_kernel<<<BB, 64, 0, stream>>>(log_startprob, log_transmat, ws);
  loss_kernel<<<NBLK_C, 256, 0, stream>>>(ws);
  final_kernel<<<1, 256, 0, stream>>>(ws, out);
}